// BondEnergyModule_49847390437978
// MI455X (gfx1250) — compile-verified
//
#include <hip/hip_runtime.h>
#include <hip/hip_bf16.h>
#include <stdint.h>

// ---------------------------------------------------------------------------
// Bond harmonic energy: gather xyz endpoints, e = par*(|d|-len)^2,
// scatter-add 0.5*e onto first atom. Memory-bound; uses CDNA5 async
// global->LDS DMA (ASYNCcnt) to stream the bond arrays, native f32 atomics
// for the scatter.
// ---------------------------------------------------------------------------

#define BLOCK 256
#define BPT   4                    // bonds per thread per tile
#define TILE  (BLOCK * BPT)        // 1024 bonds per tile
// Per-buffer LDS layout: [adj: TILE*8B][len: TILE*4B][par: TILE*4B] = 16 KB
#define ADJ_BYTES (TILE * 8)
#define LEN_OFF   (ADJ_BYTES)
#define PAR_OFF   (ADJ_BYTES + TILE * 4)
#define BUF_BYTES (ADJ_BYTES + TILE * 4 + TILE * 4)

// CDNA5 async copy: 16 bytes global -> LDS, tracked on ASYNCcnt.
__device__ __forceinline__ void async_ld16(uint32_t lds_off, const void* gaddr) {
    asm volatile("global_load_async_to_lds_b128 %0, %1, off"
                 :
                 : "v"(lds_off), "v"(gaddr)
                 : "memory");
}

__device__ __forceinline__ void wait_async0() {
    asm volatile("s_wait_asynccnt 0" ::: "memory");
}

__device__ __forceinline__ void load_f3(float dst[3], const float* __restrict__ p) {
    __builtin_memcpy(dst, p, 12);  // 4B-aligned 12B load (b96 / b64+b32)
}

// Guaranteed-native agent-scope f32 atomic add (global_atomic_add_f32).
__device__ __forceinline__ void atomic_add_f32(float* p, float v) {
    unsafeAtomicAdd(p, v);
}

__global__ void __launch_bounds__(BLOCK)
bond_energy_kernel(const float* __restrict__ xyz,
                   const int2*  __restrict__ adj,
                   const float* __restrict__ blen,
                   const float* __restrict__ bpar,
                   float*       __restrict__ out,
                   int nBonds, int numFullTiles, int tilesPerBlock)
{
    __shared__ char smem[2][BUF_BYTES];   // 32 KB double buffer

    const int tid = threadIdx.x;
    const int t0  = blockIdx.x * tilesPerBlock;
    int t1 = t0 + tilesPerBlock;
    if (t1 > numFullTiles) t1 = numFullTiles;

    uint32_t ldsBase[2];
    ldsBase[0] = (uint32_t)(uintptr_t)(&smem[0][0]);
    ldsBase[1] = (uint32_t)(uintptr_t)(&smem[1][0]);

    // ---- async stage of one full tile's bond data into LDS buffer b ----
    auto issue_tile = [&](int tile, uint32_t b) {
        const size_t s = (size_t)tile * TILE;
        const char* gadj = (const char*)(adj  + s);   // 8 KB
        const char* glen = (const char*)(blen + s);   // 4 KB
        const char* gpar = (const char*)(bpar + s);   // 4 KB
        const uint32_t o = (uint32_t)tid * 16u;
        async_ld16(b + o,                 gadj + o);
        async_ld16(b + 4096u + o,         gadj + 4096u + o);
        async_ld16(b + LEN_OFF + o,       glen + o);
        async_ld16(b + PAR_OFF + o,       gpar + o);
    };

    if (t0 < t1) issue_tile(t0, ldsBase[0]);

    for (int t = t0; t < t1; ++t) {
        const int cur = (t - t0) & 1;

        wait_async0();        // this wave's async copies for tile t are done
        __syncthreads();      // all waves' data visible; prev buffer free

        if (t + 1 < t1) issue_tile(t + 1, ldsBase[cur ^ 1]);

        const int2*  s_adj = (const int2*) (&smem[cur][0]);
        const float* s_len = (const float*)(&smem[cur][LEN_OFF]);
        const float* s_par = (const float*)(&smem[cur][PAR_OFF]);

        int2  ab[BPT];
        float L[BPT], P[BPT];
        float c0[BPT][3], c1[BPT][3];

#pragma unroll
        for (int k = 0; k < BPT; ++k) {
            const int j = (k << 8) + tid;        // k*BLOCK + tid
            ab[k] = s_adj[j];
            L[k]  = s_len[j];
            P[k]  = s_par[j];
        }
        // Issue all random gathers up front (L2-resident, latency-hidden).
#pragma unroll
        for (int k = 0; k < BPT; ++k) {
            load_f3(c0[k], xyz + 3 * (size_t)ab[k].x);
            load_f3(c1[k], xyz + 3 * (size_t)ab[k].y);
        }
#pragma unroll
        for (int k = 0; k < BPT; ++k) {
            const float dx = c0[k][0] - c1[k][0];
            const float dy = c0[k][1] - c1[k][1];
            const float dz = c0[k][2] - c1[k][2];
            const float d  = sqrtf(dx * dx + dy * dy + dz * dz);
            const float df = d - L[k];
            atomic_add_f32(&out[ab[k].x], 0.5f * P[k] * df * df);
        }
    }

    // ---- tail bonds not covered by full tiles: direct path ----
    const int tailStart = numFullTiles * TILE;
    for (int i = tailStart + blockIdx.x * BLOCK + tid; i < nBonds;
         i += gridDim.x * BLOCK) {
        const int2 ab = adj[i];
        float c0[3], c1[3];
        load_f3(c0, xyz + 3 * (size_t)ab.x);
        load_f3(c1, xyz + 3 * (size_t)ab.y);
        const float dx = c0[0] - c1[0];
        const float dy = c0[1] - c1[1];
        const float dz = c0[2] - c1[2];
        const float d  = sqrtf(dx * dx + dy * dy + dz * dz);
        const float df = d - blen[i];
        atomic_add_f32(&out[ab.x], 0.5f * bpar[i] * df * df);
    }
}

extern "C" void kernel_launch(void* const* d_in, const int* in_sizes, int n_in,
                              void* d_out, int out_size, void* d_ws, size_t ws_size,
                              hipStream_t stream) {
    const float* xyz  = (const float*)d_in[0];
    const int*   adj  = (const int*)  d_in[1];  // [E,2] int32
    const float* blen = (const float*)d_in[2];  // [E,1]
    const float* bpar = (const float*)d_in[3];  // [E,1]
    float*       out  = (float*)d_out;          // [N,1] f32

    const int nBonds = in_sizes[1] / 2;

    // Harness poisons d_out; we accumulate with atomics, so zero it first.
    hipMemsetAsync(d_out, 0, sizeof(float) * (size_t)out_size, stream);

    const int numFullTiles = nBonds / TILE;
    int grid = 1024;
    if (numFullTiles > 0 && grid > numFullTiles) grid = numFullTiles;
    if (grid < 1) grid = 1;
    const int tilesPerBlock = (numFullTiles + grid - 1) / grid;

    bond_energy_kernel<<<grid, BLOCK, 0, stream>>>(
        xyz, (const int2*)adj, blen, bpar, out,
        nBonds, numFullTiles, tilesPerBlock);
}